// LuongAttnDecoderMogLSTM_31585189494951
// MI455X (gfx1250) — compile-verified
//
#include <hip/hip_runtime.h>
#include <hip/hip_bf16.h>
#include <math.h>

// ---------------------------------------------------------------------------
// LuongAttnDecoderMogLSTM one-step decoder for MI455X (gfx1250, wave32).
// All GEMMs use V_WMMA_F32_16X16X4_F32 (fp32 in / fp32 acc -> exact vs ref).
// Bandwidth-dominated by two streams over encoder_outputs (2 x 256 MB).
// ---------------------------------------------------------------------------

typedef __attribute__((ext_vector_type(2))) float v2f;
typedef __attribute__((ext_vector_type(8))) float v8f;

constexpr int H  = 1024;
constexpr int B  = 128;
constexpr int S  = 512;
constexpr int V  = 32000;
constexpr int BH = B * H;

__device__ __forceinline__ float sigmoidf(float x) { return 1.0f / (1.0f + expf(-x)); }

__device__ __forceinline__ v8f wmma4(v2f a, v2f b, v8f c) {
  // (neg_a, A, neg_b, B, c_mod, C, reuse_a, reuse_b)
  return __builtin_amdgcn_wmma_f32_16x16x4_f32(false, a, false, b, (short)0, c,
                                               false, false);
}

__device__ __forceinline__ float wave_max(float v) {
  #pragma unroll
  for (int o = 16; o > 0; o >>= 1) v = fmaxf(v, __shfl_xor(v, o, 32));
  return v;
}
__device__ __forceinline__ float wave_sum(float v) {
  #pragma unroll
  for (int o = 16; o > 0; o >>= 1) v += __shfl_xor(v, o, 32);
  return v;
}

// ---------------------------------------------------------------------------
// Generic WMMA fp32 GEMM:  C[M,N] = epi( A1[M,K1] @ W1[N,K1]^T
//                                      (+ A2[M,K2] @ W2[N,K2]^T)
//                                      (+ bias1[N]) (+ bias2[N]) )
// One wave owns a 16 x (16*NT) output tile, marching K in steps of 4.
// Per-lane fragments (ISA 7.12.2, 32-bit layouts):
//   A: lane L -> row m=L&15, K pair at kk=(L>>4)*2           (float2 load)
//   B: lane L -> col n=L&15, K pair at kk=(L>>4)*2  (from W row-major)
//   D: acc[v] -> row v + (L>>4)*8, col L&15
// EPI: 0 = store, 1 = tanh-store, 2 = mogrifier: C *= 2*sigmoid(acc+bias)
// ---------------------------------------------------------------------------
template <int NT>
__device__ __forceinline__ void gemm_accum(v8f* acc, const float* __restrict__ A,
                                           int lda, const float* __restrict__ W,
                                           int ldw, int K, int m, int nlb, int kk) {
  const float* arow = A + (size_t)m * lda + kk;
  const float* wr[NT];
  #pragma unroll
  for (int t = 0; t < NT; ++t)
    wr[t] = W + (size_t)(nlb + 16 * t) * ldw + kk;
  #pragma unroll 4
  for (int k = 0; k < K; k += 4) {
    v2f a = *(const v2f*)(arow + k);
    #pragma unroll
    for (int t = 0; t < NT; ++t) {
      v2f b = *(const v2f*)(wr[t] + k);
      acc[t] = wmma4(a, b, acc[t]);
    }
  }
}

template <int NT, int EPI>
__launch_bounds__(256)
__global__ void wmma_gemm_kernel(const float* __restrict__ A1, int lda1,
                                 const float* __restrict__ W1, int ldw1, int K1,
                                 const float* __restrict__ A2, int lda2,
                                 const float* __restrict__ W2, int ldw2, int K2,
                                 const float* __restrict__ bias1,
                                 const float* __restrict__ bias2,
                                 float* __restrict__ C, int ldc, int M, int N) {
  const int lane   = threadIdx.x & 31;
  const int wave   = blockIdx.x * (blockDim.x >> 5) + (threadIdx.x >> 5);
  const int ntiles = N / (16 * NT);
  const int mt     = wave / ntiles;
  const int nt     = wave % ntiles;
  if (mt >= M / 16) return;  // wave-uniform

  const int m   = mt * 16 + (lane & 15);
  const int nlb = nt * (16 * NT) + (lane & 15);  // per-lane column base
  const int kk  = (lane >> 4) * 2;

  v8f acc[NT];
  #pragma unroll
  for (int t = 0; t < NT; ++t)
    #pragma unroll
    for (int i = 0; i < 8; ++i) acc[t][i] = 0.0f;

  gemm_accum<NT>(acc, A1, lda1, W1, ldw1, K1, m, nlb, kk);
  if (K2 > 0) gemm_accum<NT>(acc, A2, lda2, W2, ldw2, K2, m, nlb, kk);

  const int mrow0 = mt * 16 + (lane >> 4) * 8;
  #pragma unroll
  for (int t = 0; t < NT; ++t) {
    const int n = nlb + 16 * t;
    float bb = 0.0f;
    if (bias1) bb += bias1[n];
    if (bias2) bb += bias2[n];
    #pragma unroll
    for (int vv = 0; vv < 8; ++vv) {
      float val = acc[t][vv] + bb;
      float* p = C + (size_t)(mrow0 + vv) * ldc + n;
      if (EPI == 0)      *p = val;
      else if (EPI == 1) *p = tanhf(val);
      else               *p = *p * (2.0f * sigmoidf(val));  // mogrifier gate
    }
  }
}

// ---------------------------------------------------------------------------
// Elementwise / streaming kernels
// ---------------------------------------------------------------------------
__launch_bounds__(256)
__global__ void gather_copy_kernel(const int* __restrict__ ids,
                                   const float* __restrict__ emb,
                                   const float* __restrict__ h1,
                                   float* __restrict__ x, float* __restrict__ hb) {
  int idx = blockIdx.x * 256 + threadIdx.x;     // over B*H
  int b = idx >> 10, hh = idx & (H - 1);
  x[idx]  = emb[(size_t)ids[b] * H + hh];
  hb[idx] = h1[idx];
}

__launch_bounds__(256)
__global__ void lstm_cell_kernel(const float* __restrict__ gates,
                                 const float* __restrict__ c_in,
                                 float* __restrict__ h_out) {
  int idx = blockIdx.x * 256 + threadIdx.x;     // over B*H
  int b = idx >> 10, hh = idx & (H - 1);
  const float* row = gates + (size_t)b * 4 * H;
  float i_ = row[hh], f_ = row[H + hh], g_ = row[2 * H + hh], o_ = row[3 * H + hh];
  float c = sigmoidf(i_) * tanhf(g_);
  if (c_in) c += sigmoidf(f_) * c_in[idx];
  h_out[idx] = sigmoidf(o_) * tanhf(c);
}

// Layer-2 mogrifier with zero h-state collapses to a per-feature scale (exact).
__launch_bounds__(256)
__global__ void mog2_scale_kernel(const float* __restrict__ h1n,
                                  const float* __restrict__ mog2_b,
                                  float* __restrict__ x2) {
  int idx = blockIdx.x * 256 + threadIdx.x;
  int hh = idx & (H - 1);
  float g = (2.0f * sigmoidf(mog2_b[hh])) *
            (2.0f * sigmoidf(mog2_b[2 * H + hh])) *
            (2.0f * sigmoidf(mog2_b[4 * H + hh]));
  x2[idx] = h1n[idx] * g;
}

// scores[b,s] = sum_h rnn[b,h] * enc[s,b,h]   (grid: (B, S/64), block 256)
__launch_bounds__(256)
__global__ void attn_scores_kernel(const float* __restrict__ rnn,
                                   const float* __restrict__ enc,
                                   float* __restrict__ scores) {
  const int b = blockIdx.x;
  const int lane = threadIdx.x & 31, wid = threadIdx.x >> 5;
  float4 r[8];
  const float4* rp = (const float4*)(rnn + (size_t)b * H);
  #pragma unroll
  for (int i = 0; i < 8; ++i) r[i] = rp[i * 32 + lane];
  #pragma unroll
  for (int j = 0; j < 8; ++j) {
    int s = blockIdx.y * 64 + wid * 8 + j;
    const float4* ep = (const float4*)(enc + ((size_t)s * B + b) * H);
    float acc = 0.0f;
    #pragma unroll
    for (int i = 0; i < 8; ++i) {
      float4 e = ep[i * 32 + lane];
      acc += r[i].x * e.x + r[i].y * e.y + r[i].z * e.z + r[i].w * e.w;
    }
    acc = wave_sum(acc);
    if (lane == 0) scores[(size_t)b * S + s] = acc;
  }
}

__launch_bounds__(256)
__global__ void softmax_s_kernel(float* __restrict__ scores) {
  __shared__ float red[8];
  int b = blockIdx.x, tid = threadIdx.x, lane = tid & 31, wid = tid >> 5;
  float* row = scores + (size_t)b * S;
  float v0 = row[tid], v1 = row[tid + 256];
  float m = wave_max(fmaxf(v0, v1));
  if (lane == 0) red[wid] = m;
  __syncthreads();
  float bm = red[0];
  #pragma unroll
  for (int i = 1; i < 8; ++i) bm = fmaxf(bm, red[i]);
  __syncthreads();
  float e0 = expf(v0 - bm), e1 = expf(v1 - bm);
  float s = wave_sum(e0 + e1);
  if (lane == 0) red[wid] = s;
  __syncthreads();
  float bs = 0.0f;
  #pragma unroll
  for (int i = 0; i < 8; ++i) bs += red[i];
  float inv = 1.0f / bs;
  row[tid] = e0 * inv;
  row[tid + 256] = e1 * inv;
}

// ctx[b,h] = sum_s attn[b,s] * enc[s,b,h]   (grid: B, block 256, float4/thread)
__launch_bounds__(256)
__global__ void context_kernel(const float* __restrict__ attn,
                               const float* __restrict__ enc,
                               float* __restrict__ ctx) {
  __shared__ float a[S];
  int b = blockIdx.x, tid = threadIdx.x;
  a[tid] = attn[(size_t)b * S + tid];
  a[tid + 256] = attn[(size_t)b * S + tid + 256];
  __syncthreads();
  float4 acc = make_float4(0.f, 0.f, 0.f, 0.f);
  for (int s = 0; s < S; ++s) {
    float4 e = *(const float4*)(enc + ((size_t)s * B + b) * H + tid * 4);
    float w = a[s];
    acc.x += w * e.x; acc.y += w * e.y; acc.z += w * e.z; acc.w += w * e.w;
  }
  *(float4*)(ctx + (size_t)b * H + tid * 4) = acc;
}

__launch_bounds__(1024)
__global__ void softmax_v_kernel(float* __restrict__ out) {
  __shared__ float red[32];
  int b = blockIdx.x, tid = threadIdx.x, lane = tid & 31, wid = tid >> 5;
  float* row = out + (size_t)b * V;
  float m = -3.4e38f;
  for (int i = tid; i < V; i += 1024) m = fmaxf(m, row[i]);
  m = wave_max(m);
  if (lane == 0) red[wid] = m;
  __syncthreads();
  float bm = red[0];
  #pragma unroll
  for (int i = 1; i < 32; ++i) bm = fmaxf(bm, red[i]);
  __syncthreads();
  float s = 0.0f;
  for (int i = tid; i < V; i += 1024) { float e = expf(row[i] - bm); row[i] = e; s += e; }
  s = wave_sum(s);
  if (lane == 0) red[wid] = s;
  __syncthreads();
  float bs = 0.0f;
  #pragma unroll
  for (int i = 0; i < 32; ++i) bs += red[i];
  float inv = 1.0f / bs;
  for (int i = tid; i < V; i += 1024) row[i] *= inv;
}

// ---------------------------------------------------------------------------
extern "C" void kernel_launch(void* const* d_in, const int* in_sizes, int n_in,
                              void* d_out, int out_size, void* d_ws, size_t ws_size,
                              hipStream_t stream) {
  const int*   ids      = (const int*)  d_in[0];
  const float* enc      = (const float*)d_in[1];
  const float* h1       = (const float*)d_in[2];
  const float* c1       = (const float*)d_in[3];
  const float* emb      = (const float*)d_in[4];
  const float* mog1_W   = (const float*)d_in[5];
  const float* mog1_b   = (const float*)d_in[6];
  const float* l1_Wih   = (const float*)d_in[7];
  const float* l1_Whh   = (const float*)d_in[8];
  const float* l1_bih   = (const float*)d_in[9];
  const float* l1_bhh   = (const float*)d_in[10];
  // mog2_W (d_in[11]) unused: layer-2 mogrifier matmuls vanish for zero state
  const float* mog2_b   = (const float*)d_in[12];
  const float* l2_Wih   = (const float*)d_in[13];
  // l2_Whh (d_in[14]) unused: h-state is zero
  const float* l2_bih   = (const float*)d_in[15];
  const float* l2_bhh   = (const float*)d_in[16];
  const float* fc_W     = (const float*)d_in[17];
  const float* fc_b     = (const float*)d_in[18];
  const float* concat_W = (const float*)d_in[19];
  const float* concat_b = (const float*)d_in[20];
  const float* out_W    = (const float*)d_in[21];
  const float* out_b    = (const float*)d_in[22];

  float* out    = (float*)d_out;           // [B, V] softmax output
  float* h2     = out + (size_t)B * V;     // [1, B, H] hidden output (tuple #2)

  float* ws     = (float*)d_ws;
  float* x      = ws;                      // BH
  float* hb     = x + BH;                  // BH (mutable copy of h1)
  float* gates  = hb + BH;                 // 4*BH
  float* h1n    = gates + 4 * BH;          // BH
  float* x2     = h1n + BH;                // BH
  float* rnn    = x2 + BH;                 // BH
  float* scores = rnn + BH;                // B*S (softmax in place -> attn)
  float* ctx    = scores + B * S;          // BH
  float* cc     = ctx + BH;                // BH (concat_out)

  // 1) embedding gather + copy h1 into mutable buffer
  gather_copy_kernel<<<BH / 256, 256, 0, stream>>>(ids, emb, h1, x, hb);

  // 2) layer-1 mogrifier: 5 alternating gated updates, each a 128x1024x1024 GEMM
  //    odd step (i=0,2,4): x *= 2*sig(h @ W_i^T + b_i); even: h *= 2*sig(x @ ...)
  for (int i = 0; i < 5; ++i) {
    const float* src = (i & 1) ? x : hb;
    float*       tgt = (i & 1) ? hb : x;
    wmma_gemm_kernel<2, 2><<<32, 256, 0, stream>>>(
        src, H, mog1_W + (size_t)i * H * H, H, H,
        nullptr, 0, nullptr, 0, 0,
        mog1_b + (size_t)i * H, nullptr, tgt, H, B, H);
  }

  // 3) layer-1 gates = x@Wih^T + h@Whh^T + bih + bhh   [128, 4096]
  wmma_gemm_kernel<2, 0><<<128, 256, 0, stream>>>(
      x, H, l1_Wih, H, H, hb, H, l1_Whh, H, H,
      l1_bih, l1_bhh, gates, 4 * H, B, 4 * H);

  // 4) layer-1 LSTM cell -> h1n
  lstm_cell_kernel<<<BH / 256, 256, 0, stream>>>(gates, c1, h1n);

  // 5) layer-2 mogrifier (zero state) -> per-feature scale of h1n
  mog2_scale_kernel<<<BH / 256, 256, 0, stream>>>(h1n, mog2_b, x2);

  // 6) layer-2 gates = x2@Wih^T + bih + bhh (Whh term vanishes)
  wmma_gemm_kernel<2, 0><<<128, 256, 0, stream>>>(
      x2, H, l2_Wih, H, H, nullptr, 0, nullptr, 0, 0,
      l2_bih, l2_bhh, gates, 4 * H, B, 4 * H);

  // 7) layer-2 LSTM cell (c=0) -> h2 (written straight into d_out tail = hidden)
  lstm_cell_kernel<<<BH / 256, 256, 0, stream>>>(gates, nullptr, h2);

  // 8) rnn_out = h2 @ fc_W^T + fc_b
  wmma_gemm_kernel<2, 0><<<32, 256, 0, stream>>>(
      h2, H, fc_W, H, H, nullptr, 0, nullptr, 0, 0,
      fc_b, nullptr, rnn, H, B, H);

  // 9) attention: scores -> softmax -> context (two streams over 256 MB enc)
  attn_scores_kernel<<<dim3(B, S / 64), 256, 0, stream>>>(rnn, enc, scores);
  softmax_s_kernel<<<B, 256, 0, stream>>>(scores);
  context_kernel<<<B, 256, 0, stream>>>(scores, enc, ctx);

  // 10) concat_out = tanh([rnn, ctx] @ concat_W^T + b)  (K split over two A's)
  wmma_gemm_kernel<2, 1><<<32, 256, 0, stream>>>(
      rnn, H, concat_W, 2 * H, H, ctx, H, concat_W + H, 2 * H, H,
      concat_b, nullptr, cc, H, B, H);

  // 11) logits = cc @ out_W^T + out_b  -> d_out, then softmax in place
  wmma_gemm_kernel<4, 0><<<500, 256, 0, stream>>>(
      cc, H, out_W, H, H, nullptr, 0, nullptr, 0, 0,
      out_b, nullptr, out, V, B, V);
  softmax_v_kernel<<<B, 1024, 0, stream>>>(out);
}